// LinerConv3DLayer_12257836663029
// MI455X (gfx1250) — compile-verified
//
#include <hip/hip_runtime.h>

// Problem constants (from reference config)
#define B_    4
#define F_    4
#define C_    4
#define DK    3
#define D_    16
#define DOUT  14          // D - DK + 1
#define H_    128
#define W_    128
#define HW    (H_ * W_)
#define ALPHA 0.2f

typedef __attribute__((ext_vector_type(2))) float v2f;
typedef __attribute__((ext_vector_type(8))) float v8f;

// ---------------------------------------------------------------------------
// Stage 1: per-pixel contraction via V_WMMA_F32_16X16X4_F32 (exact fp32).
//   q[b,f,i,pix] = sum_{c,dd} x[b,c,i+dd,pix] * w[f,c,dd,pix] + sum_{c,dd} bias[f,c,dd,pix]
// GEMM view per pixel: M=(b,i) rows (56, in 4 tiles of 16), K=c (3 chunks, one
// per dd), N=f (4 of 16 cols). Accumulator initialized with the bias sum.
// One wave32 per pixel; EXEC is all-ones at every WMMA (selects, not branches).
// ---------------------------------------------------------------------------
__global__ __launch_bounds__(256)
void qstage_wmma(const float* __restrict__ x, const float* __restrict__ w,
                 const float* __restrict__ bias, float* __restrict__ q) {
  int wave = (int)((blockIdx.x * blockDim.x + threadIdx.x) >> 5);
  int lane = (int)(threadIdx.x & 31);
  if (wave >= HW) return;                 // uniform per wave
  const int pix   = wave;
  const int n     = lane & 15;            // N (= f) column this lane owns
  const int half  = lane >> 4;            // 0: K∈{0,1}; 1: K∈{2,3}
  const int khalf = half << 1;            // base c index for this lane half

  // --- B matrices (weights), one 4x16 per dd chunk. B[k=c][n=f].
  // Layout: VGPR0 -> K=khalf, VGPR1 -> K=khalf+1, N = lane&15.
  const int fcl = (n < F_) ? n : (F_ - 1);           // clamped safe address
  v2f Bm[DK];
  #pragma unroll
  for (int dd = 0; dd < DK; ++dd) {
    const float* wp = w + (size_t)((fcl * C_) * DK + dd) * HW + pix;
    float b0 = wp[(size_t)((khalf + 0) * DK) * HW];  // + c*DK*HW
    float b1 = wp[(size_t)((khalf + 1) * DK) * HW];
    Bm[dd].x = (n < F_) ? b0 : 0.0f;
    Bm[dd].y = (n < F_) ? b1 : 0.0f;
  }

  // --- Bias sum per column f (same value down every row of C).
  float bs = 0.0f;
  #pragma unroll
  for (int c = 0; c < C_; ++c)
    #pragma unroll
    for (int dd = 0; dd < DK; ++dd)
      bs += bias[(size_t)((fcl * C_ + c) * DK + dd) * HW + pix];
  const float cinit = (n < F_) ? bs : 0.0f;

  // --- 4 M-tiles of 16 rows; rows 56..63 of the last tile are zeroed in A.
  for (int t = 0; t < 4; ++t) {
    v8f acc;
    #pragma unroll
    for (int j = 0; j < 8; ++j) acc[j] = cinit;

    #pragma unroll
    for (int dd = 0; dd < DK; ++dd) {
      int m  = t * 16 + n;                 // A row this lane supplies
      int mc = (m < B_ * DOUT) ? m : (B_ * DOUT - 1);
      int bb = mc / DOUT, ii = mc % DOUT;
      const float* xp = x + (size_t)((bb * C_) * D_ + (ii + dd)) * HW + pix;
      float a0 = xp[(size_t)((khalf + 0) * D_) * HW];   // + c*D*HW
      float a1 = xp[(size_t)((khalf + 1) * D_) * HW];
      v2f A;
      A.x = (m < B_ * DOUT) ? a0 : 0.0f;
      A.y = (m < B_ * DOUT) ? a1 : 0.0f;
      acc = __builtin_amdgcn_wmma_f32_16x16x4_f32(false, A, false, Bm[dd],
                                                  (short)0, acc, false, false);
    }

    // Store: C/D layout — VGPR j holds row m = t*16 + j + 8*half, col n.
    #pragma unroll
    for (int j = 0; j < 8; ++j) {
      int m = t * 16 + j + (half << 3);
      if (m < B_ * DOUT && n < F_) {
        int bb = m / DOUT, ii = m % DOUT;
        q[(size_t)((bb * F_ + n) * DOUT + ii) * HW + pix] = acc[j];
      }
    }
  }
}

// ---------------------------------------------------------------------------
// Stage 2: zero-padded 3x3 box sum over (h,w) + leaky_relu(0.2).
// q and out share the identical [B,F,Dout,H,W] layout. Pure L2-bandwidth.
// ---------------------------------------------------------------------------
__global__ __launch_bounds__(256)
void box3_leaky(const float* __restrict__ q, float* __restrict__ out) {
  size_t idx = (size_t)blockIdx.x * blockDim.x + threadIdx.x;
  const size_t total = (size_t)B_ * F_ * DOUT * HW;
  if (idx >= total) return;
  int pix = (int)(idx % HW);
  size_t plane = idx / HW;
  int hh = pix / W_, ww = pix % W_;
  const float* qp = q + plane * HW;
  float s = 0.0f;
  #pragma unroll
  for (int dh = -1; dh <= 1; ++dh) {
    int h2 = hh + dh;
    if ((unsigned)h2 >= (unsigned)H_) continue;
    #pragma unroll
    for (int dw = -1; dw <= 1; ++dw) {
      int w2 = ww + dw;
      if ((unsigned)w2 >= (unsigned)W_) continue;
      s += qp[h2 * W_ + w2];
    }
  }
  out[idx] = (s >= 0.0f) ? s : ALPHA * s;
}

// ---------------------------------------------------------------------------
// Fallback (only if d_ws is too small for the 14.7 MB intermediate):
// fully fused direct evaluation, one thread per output.
// ---------------------------------------------------------------------------
__global__ __launch_bounds__(256)
void fused_direct(const float* __restrict__ x, const float* __restrict__ w,
                  const float* __restrict__ bias, float* __restrict__ out) {
  size_t idx = (size_t)blockIdx.x * blockDim.x + threadIdx.x;
  const size_t total = (size_t)B_ * F_ * DOUT * HW;
  if (idx >= total) return;
  int pix = (int)(idx % HW);
  size_t plane = idx / HW;
  int ii = (int)(plane % DOUT);
  int ff = (int)((plane / DOUT) % F_);
  int bb = (int)(plane / ((size_t)DOUT * F_));
  int hh = pix / W_, ww = pix % W_;
  float s = 0.0f;
  for (int dh = -1; dh <= 1; ++dh) {
    int h2 = hh + dh;
    if ((unsigned)h2 >= (unsigned)H_) continue;
    for (int dw = -1; dw <= 1; ++dw) {
      int w2 = ww + dw;
      if ((unsigned)w2 >= (unsigned)W_) continue;
      int p2 = h2 * W_ + w2;
      #pragma unroll
      for (int c = 0; c < C_; ++c)
        #pragma unroll
        for (int dd = 0; dd < DK; ++dd) {
          float xv = x[(size_t)((bb * C_ + c) * D_ + (ii + dd)) * HW + p2];
          float wv = w[(size_t)((ff * C_ + c) * DK + dd) * HW + p2];
          float bv = bias[(size_t)((ff * C_ + c) * DK + dd) * HW + p2];
          s += xv * wv + bv;
        }
    }
  }
  out[idx] = (s >= 0.0f) ? s : ALPHA * s;
}

extern "C" void kernel_launch(void* const* d_in, const int* in_sizes, int n_in,
                              void* d_out, int out_size, void* d_ws, size_t ws_size,
                              hipStream_t stream) {
  const float* x  = (const float*)d_in[0];
  const float* w  = (const float*)d_in[1];
  const float* bi = (const float*)d_in[2];
  float* out = (float*)d_out;

  const size_t total  = (size_t)B_ * F_ * DOUT * HW;   // 3,670,016
  const size_t qbytes = total * sizeof(float);         // ~14.7 MB

  if (ws_size >= qbytes) {
    float* q = (float*)d_ws;
    // Stage 1: one wave32 per pixel -> HW waves.
    const int threads1 = HW * 32;                      // 524,288
    dim3 grid1((threads1 + 255) / 256), blk(256);
    qstage_wmma<<<grid1, blk, 0, stream>>>(x, w, bi, q);
    // Stage 2: one thread per output element.
    dim3 grid2((unsigned)((total + 255) / 256));
    box3_leaky<<<grid2, blk, 0, stream>>>(q, out);
  } else {
    dim3 grid((unsigned)((total + 255) / 256)), blk(256);
    fused_direct<<<grid, blk, 0, stream>>>(x, w, bi, out);
  }
}